// SubgraphGNNWithReadout_46591805227285
// MI455X (gfx1250) — compile-verified
//
#include <hip/hip_runtime.h>
#include <hip/hip_bf16.h>
#include <math.h>

typedef __attribute__((ext_vector_type(2))) float v2f;
typedef __attribute__((ext_vector_type(8))) float v8f;

#define NN     40000
#define NE     640000
#define HIDN   128
#define NSUB   4096
#define KSUB   32
#define NCLS   10
#define BN_EPS 1e-5f

// ---------------- degree / normalization precompute ----------------

__global__ void k_deg_init(float* __restrict__ deg) {
  int i = blockIdx.x * blockDim.x + threadIdx.x;
  if (i < NN) deg[i] = 1.0f;  // self-loop weight
}

__global__ void k_deg_edges(float* __restrict__ deg, const int* __restrict__ dst,
                            const float* __restrict__ w) {
  int e = blockIdx.x * blockDim.x + threadIdx.x;
  if (e < NE) atomicAdd(&deg[dst[e]], w[e]);
}

__global__ void k_dinv(float* __restrict__ deg) {
  int i = blockIdx.x * blockDim.x + threadIdx.x;
  if (i < NN) {
    float d = deg[i];
    deg[i] = d > 0.0f ? rsqrtf(d) : 0.0f;
  }
}

__global__ void k_norm(float* __restrict__ norm, const float* __restrict__ dinv,
                       const int* __restrict__ src, const int* __restrict__ dst,
                       const float* __restrict__ w) {
  int e = blockIdx.x * blockDim.x + threadIdx.x;
  if (e < NE) norm[e] = dinv[src[e]] * w[e] * dinv[dst[e]];
}

// ---------------- WMMA f32 GEMM: C[M x N] = A[M x K] @ B[K x N] (+bias, relu) ----
// grid.x = M/64 (M multiple of 64); blockDim.x = 32 * (N/16), N <= 128, K mult of 4.
// Each wave computes FOUR 16x16 output tiles (rows m, m+16, m+32, m+48) so one
// B fragment feeds 4 x V_WMMA_F32_16X16X4_F32 (4x less B/L2 traffic, 4x wmma
// density per address calc). Dynamic LDS: 64*K floats (A row-strip).

__global__ void k_wmma_gemm(const float* __restrict__ A, const float* __restrict__ B,
                            const float* __restrict__ bias, float* __restrict__ C,
                            int N, int K, int relu) {
  extern __shared__ float As[];  // [64][K]
  const int row0 = blockIdx.x * 64;

  for (int t = threadIdx.x; t < 64 * K; t += blockDim.x) {
    int r = t / K, c = t - r * K;
    As[t] = A[(row0 + r) * K + c];
  }
  __syncthreads();

  const int lane = threadIdx.x & 31;
  const int wave = threadIdx.x >> 5;
  const int m    = lane & 15;       // row within tile (A) / col (B,C)
  const int half = lane >> 4;       // hi/lo half of wave
  const int n    = wave * 16 + m;   // output column

  v8f acc0 = {}, acc1 = {}, acc2 = {}, acc3 = {};
  #pragma unroll 4
  for (int k = 0; k < K; k += 4) {
    const int kk = k + 2 * half;
    // B 4x16: VGPR j, lane-half h -> K = k + 2h + j, col = lane&15
    v2f b;
    b[0] = B[kk * N + n];
    b[1] = B[(kk + 1) * N + n];
    // A 16x4 per tile: lanes 0-15 -> K = k+0,k+1 ; lanes 16-31 -> K = k+2,k+3
    v2f a0, a1, a2, a3;
    a0[0] = As[(m)      * K + kk]; a0[1] = As[(m)      * K + kk + 1];
    a1[0] = As[(m + 16) * K + kk]; a1[1] = As[(m + 16) * K + kk + 1];
    a2[0] = As[(m + 32) * K + kk]; a2[1] = As[(m + 32) * K + kk + 1];
    a3[0] = As[(m + 48) * K + kk]; a3[1] = As[(m + 48) * K + kk + 1];
    acc0 = __builtin_amdgcn_wmma_f32_16x16x4_f32(false, a0, false, b, (short)0, acc0, false, false);
    acc1 = __builtin_amdgcn_wmma_f32_16x16x4_f32(false, a1, false, b, (short)0, acc1, false, false);
    acc2 = __builtin_amdgcn_wmma_f32_16x16x4_f32(false, a2, false, b, (short)0, acc2, false, false);
    acc3 = __builtin_amdgcn_wmma_f32_16x16x4_f32(false, a3, false, b, (short)0, acc3, false, false);
  }

  const float bv = bias ? bias[n] : 0.0f;
  #pragma unroll
  for (int j = 0; j < 8; ++j) {
    const int r = row0 + half * 8 + j;  // C/D: VGPR j holds M = j + 8*half
    float v0 = acc0[j] + bv;
    float v1 = acc1[j] + bv;
    float v2 = acc2[j] + bv;
    float v3 = acc3[j] + bv;
    if (relu) {
      v0 = fmaxf(v0, 0.0f); v1 = fmaxf(v1, 0.0f);
      v2 = fmaxf(v2, 0.0f); v3 = fmaxf(v3, 0.0f);
    }
    C[(r)      * N + n] = v0;
    C[(r + 16) * N + n] = v1;
    C[(r + 32) * N + n] = v2;
    C[(r + 48) * N + n] = v3;
  }
}

// ---------------- GCN aggregation ----------------

__global__ void k_agg_init(float* __restrict__ agg, const float* __restrict__ hw,
                           const float* __restrict__ dinv) {
  int idx = blockIdx.x * blockDim.x + threadIdx.x;
  if (idx < NN * HIDN) {
    float di = dinv[idx >> 7];
    agg[idx] = di * di * hw[idx];  // self-loop term: dinv[i]*1*dinv[i]*hw[i]
  }
}

// one wave per edge; lane covers 4 features (32*4 = 128)
__global__ void k_agg_edges(float* __restrict__ agg, const float* __restrict__ hw,
                            const float* __restrict__ norm, const int* __restrict__ src,
                            const int* __restrict__ dst) {
  int gid  = blockIdx.x * blockDim.x + threadIdx.x;
  int e    = gid >> 5;
  int lane = gid & 31;
  if (e < NE) {
    int   s  = src[e];
    int   d  = dst[e];
    float nw = norm[e];
    const float4* hs = (const float4*)(hw + (size_t)s * HIDN);
    float4 v = hs[lane];
    float* ad = agg + (size_t)d * HIDN + lane * 4;
    atomicAdd(ad + 0, nw * v.x);
    atomicAdd(ad + 1, nw * v.y);
    atomicAdd(ad + 2, nw * v.z);
    atomicAdd(ad + 3, nw * v.w);
  }
}

// bias + BatchNorm(eval) + ReLU
__global__ void k_post(float* __restrict__ hout, const float* __restrict__ agg,
                       const float* __restrict__ bsl, const float* __restrict__ gam,
                       const float* __restrict__ bet, const float* __restrict__ mean,
                       const float* __restrict__ var) {
  int idx = blockIdx.x * blockDim.x + threadIdx.x;
  if (idx < NN * HIDN) {
    int f = idx & (HIDN - 1);
    float v = agg[idx] + bsl[f];
    v = (v - mean[f]) * rsqrtf(var[f] + BN_EPS) * gam[f] + bet[f];
    hout[idx] = fmaxf(v, 0.0f);
  }
}

// ---------------- ragged subgraph readout: [NSUB, 3*HIDN] ----------------
// one block (128 threads) per subgraph; thread = feature
__global__ void k_readout(float* __restrict__ g, const float* __restrict__ h,
                          const int* __restrict__ sub) {
  int s = blockIdx.x;
  int f = threadIdx.x;
  float sum = 0.0f, mx = -INFINITY;
  int cnt = 0;
  #pragma unroll 4
  for (int k = 0; k < KSUB; ++k) {
    int id = sub[s * KSUB + k];
    if (id >= 0) {
      float v = h[(size_t)id * HIDN + f];
      sum += v;
      mx = fmaxf(mx, v);
      ++cnt;
    }
  }
  float mean = sum / fmaxf((float)cnt, 1.0f);
  float mxo  = cnt > 0 ? mx : 0.0f;
  g[(size_t)s * (3 * HIDN) + f]            = mean;
  g[(size_t)s * (3 * HIDN) + HIDN + f]     = mxo;
  g[(size_t)s * (3 * HIDN) + 2 * HIDN + f] = sum;
}

// ---------------- final 64 -> 10 layer (too narrow for a 16-wide tile) -------
__global__ void k_mlp3(float* __restrict__ out, const float* __restrict__ z2,
                       const float* __restrict__ W, const float* __restrict__ b) {
  int idx = blockIdx.x * blockDim.x + threadIdx.x;
  if (idx < NSUB * NCLS) {
    int s = idx / NCLS, c = idx - s * NCLS;
    float acc = b[c];
    #pragma unroll
    for (int k = 0; k < 64; ++k) acc += z2[s * 64 + k] * W[k * NCLS + c];
    out[idx] = acc;
  }
}

// ---------------- driver ----------------

extern "C" void kernel_launch(void* const* d_in, const int* in_sizes, int n_in,
                              void* d_out, int out_size, void* d_ws, size_t ws_size,
                              hipStream_t stream) {
  const float* x   = (const float*)d_in[0];
  const int*   ei  = (const int*)d_in[1];       // [2, NE]: src row then dst row
  const float* ea  = (const float*)d_in[2];
  const int*   sub = (const int*)d_in[3];
  const float* Ws  = (const float*)d_in[4];     // [3, 128, 128]
  const float* bs  = (const float*)d_in[5];
  const float* bgm = (const float*)d_in[6];
  const float* bbt = (const float*)d_in[7];
  const float* bmn = (const float*)d_in[8];
  const float* bvr = (const float*)d_in[9];
  const float* Wc1 = (const float*)d_in[10];    // [384, 128]
  const float* bc1 = (const float*)d_in[11];
  const float* Wc2 = (const float*)d_in[12];    // [128, 64]
  const float* bc2 = (const float*)d_in[13];
  const float* Wc3 = (const float*)d_in[14];    // [64, 10]
  const float* bc3 = (const float*)d_in[15];
  float* out = (float*)d_out;

  const int* src = ei;
  const int* dst = ei + NE;

  // workspace layout (~42 MB)
  float* dinv = (float*)d_ws;             // 40960 (padded)
  float* norm = dinv + 40960;             // 655360 (padded)
  float* buf0 = norm + 655360;            // NN*HIDN = 5,120,000
  float* buf1 = buf0 + (size_t)NN * HIDN; // NN*HIDN
  // g/z1/z2 overlay buf1 (free after the 3rd layer; final h lands in buf0)
  float* g  = buf1;
  float* z1 = g + (size_t)NSUB * 3 * HIDN;
  float* z2 = z1 + (size_t)NSUB * HIDN;

  // --- gcn_norm ---
  k_deg_init<<<(NN + 255) / 256, 256, 0, stream>>>(dinv);
  k_deg_edges<<<(NE + 255) / 256, 256, 0, stream>>>(dinv, dst, ea);
  k_dinv<<<(NN + 255) / 256, 256, 0, stream>>>(dinv);
  k_norm<<<(NE + 255) / 256, 256, 0, stream>>>(norm, dinv, src, dst, ea);

  // --- 3 GCN layers (ping-pong buf0/buf1) ---
  const float* cur = x;
  float* bA = buf0;  // gemm output (hw), then reused for h_next
  float* bB = buf1;  // aggregation accumulator
  for (int l = 0; l < 3; ++l) {
    k_wmma_gemm<<<dim3(NN / 64, 1), 256, 64 * HIDN * sizeof(float), stream>>>(
        cur, Ws + (size_t)l * HIDN * HIDN, nullptr, bA, HIDN, HIDN, 0);
    k_agg_init<<<(NN * HIDN + 255) / 256, 256, 0, stream>>>(bB, bA, dinv);
    k_agg_edges<<<(NE * 32 + 255) / 256, 256, 0, stream>>>(bB, bA, norm, src, dst);
    k_post<<<(NN * HIDN + 255) / 256, 256, 0, stream>>>(
        bA, bB, bs + l * HIDN, bgm + l * HIDN, bbt + l * HIDN,
        bmn + l * HIDN, bvr + l * HIDN);
    cur = bA;
    float* t = bA; bA = bB; bB = t;
  }
  // after 3 layers cur == buf0, buf1 free

  // --- readout + classifier ---
  k_readout<<<NSUB, HIDN, 0, stream>>>(g, cur, sub);
  k_wmma_gemm<<<dim3(NSUB / 64, 1), 256, 64 * 384 * sizeof(float), stream>>>(
      g, Wc1, bc1, z1, HIDN, 3 * HIDN, 1);
  k_wmma_gemm<<<dim3(NSUB / 64, 1), 128, 64 * HIDN * sizeof(float), stream>>>(
      z1, Wc2, bc2, z2, 64, HIDN, 1);
  k_mlp3<<<(NSUB * NCLS + 255) / 256, 256, 0, stream>>>(out, z2, Wc3, bc3);
}